// Net_10136122819212
// MI455X (gfx1250) — compile-verified
//
#include <hip/hip_runtime.h>
#include <hip/hip_bf16.h>

#define IN_DIM 512
#define HID    16
#define OUTC   40

typedef float v2f __attribute__((ext_vector_type(2)));
typedef float v8f __attribute__((ext_vector_type(8)));

// ---------------------------------------------------------------------------
// dst[i] = bias[i % dim]   (initialize aggregation buffers with the bias)
// ---------------------------------------------------------------------------
__global__ void k_init_bias(float* __restrict__ dst, const float* __restrict__ bias,
                            int dim, long long total) {
  long long t = (long long)blockIdx.x * blockDim.x + threadIdx.x;
  if (t < total) dst[t] = bias[(int)(t % dim)];
}

// ---------------------------------------------------------------------------
// GEMM1: h1pre[100000,16] = x[100000,512] @ W1[512,16], fp32 WMMA 16x16x4.
// One wave per 16-row tile (100000 = 6250 * 16, exact).
// Lane layout (wave32): m = lane&15, hi = lane>>4.
//   A vgpr r : x[row=tile*16+m][4k + r + 2*hi]
//   B vgpr r : W1[4k + r + 2*hi][m]
//   D vgpr r : row (r + 8*hi), col m
// ---------------------------------------------------------------------------
__global__ void k_gemm1_wmma(const float* __restrict__ x, const float* __restrict__ W1,
                             float* __restrict__ h1, int ntiles) {
  int wave = (int)(((long long)blockIdx.x * blockDim.x + threadIdx.x) >> 5);
  if (wave >= ntiles) return;            // wave-uniform: EXEC stays all-ones
  int lane = threadIdx.x & 31;
  int m  = lane & 15;
  int hi = lane >> 4;

  const float* xr = x  + (size_t)(wave * 16 + m) * IN_DIM + 2 * hi;
  const float* wr = W1 + (size_t)(2 * hi) * HID + m;

  v8f acc = {};
#pragma unroll 8
  for (int k = 0; k < IN_DIM; k += 4) {
    v2f a, b;
    a.x = xr[k];                 // x[row][k + 2*hi]
    a.y = xr[k + 1];             // x[row][k + 2*hi + 1]
    b.x = wr[k * HID];           // W1[k + 2*hi][m]
    b.y = wr[k * HID + HID];     // W1[k + 2*hi + 1][m]
    acc = __builtin_amdgcn_wmma_f32_16x16x4_f32(
        /*neg_a=*/false, a, /*neg_b=*/false, b,
        /*c_mod=*/(short)0, acc, /*reuse_a=*/false, /*reuse_b=*/false);
  }

  float* out = h1 + (size_t)wave * 16 * HID + (size_t)(8 * hi) * HID + m;
#pragma unroll
  for (int r = 0; r < 8; ++r) out[r * HID] = acc[r];
}

// ---------------------------------------------------------------------------
// scatter1: agg1[dst[e]][c] += h1pre[src[e]][c], 16 lanes per edge (coalesced
// 64B gather + 16 contiguous fp32 atomics; agg1/h1pre are L2 resident).
// ---------------------------------------------------------------------------
__global__ void k_scatter16(const float* __restrict__ feat, float* __restrict__ agg,
                            const int* __restrict__ edges, long long E) {
  long long t = (long long)blockIdx.x * blockDim.x + threadIdx.x;
  if (t >= E * HID) return;
  long long e = t >> 4;
  int c = (int)(t & 15);
  int s = edges[e];
  int d = edges[E + e];
  unsafeAtomicAdd(&agg[(size_t)d * HID + c], feat[(size_t)s * HID + c]);
}

// ---------------------------------------------------------------------------
// h2pre[i][j] = sum_k relu(agg1[i][k]) * W2[k][j]   (K=16, N=40: VALU FMA)
// thread per (i,j): coalesced writes; W2 column loads coalesce across lanes.
// ---------------------------------------------------------------------------
__global__ void k_relu_gemm2(const float* __restrict__ agg1, const float* __restrict__ W2,
                             float* __restrict__ h2, long long N) {
  long long t = (long long)blockIdx.x * blockDim.x + threadIdx.x;
  if (t >= N * OUTC) return;
  long long i = t / OUTC;
  int j = (int)(t - i * OUTC);
  const float* row = agg1 + (size_t)i * HID;
  float s = 0.f;
#pragma unroll
  for (int k = 0; k < HID; ++k) {
    float v = row[k];
    v = v > 0.f ? v : 0.f;
    s = fmaf(v, W2[k * OUTC + j], s);
  }
  h2[t] = s;
}

// ---------------------------------------------------------------------------
// scatter2: out[dst[e]][c] += h2pre[src[e]][c], 40 lanes per edge.
// ---------------------------------------------------------------------------
__global__ void k_scatter40(const float* __restrict__ feat, float* __restrict__ out,
                            const int* __restrict__ edges, long long E) {
  long long t = (long long)blockIdx.x * blockDim.x + threadIdx.x;
  if (t >= E * OUTC) return;
  long long e = t / OUTC;
  int c = (int)(t - e * OUTC);
  int s = edges[e];
  int d = edges[E + e];
  unsafeAtomicAdd(&out[(size_t)d * OUTC + c], feat[(size_t)s * OUTC + c]);
}

// ---------------------------------------------------------------------------
// In-place row-wise log_softmax over 40 classes; one thread per row.
// ---------------------------------------------------------------------------
__global__ void k_log_softmax(float* __restrict__ out, long long N) {
  long long i = (long long)blockIdx.x * blockDim.x + threadIdx.x;
  if (i >= N) return;
  float* row = out + (size_t)i * OUTC;
  float r[OUTC];
  float mx = -INFINITY;
#pragma unroll
  for (int j = 0; j < OUTC; ++j) { r[j] = row[j]; mx = fmaxf(mx, r[j]); }
  float s = 0.f;
#pragma unroll
  for (int j = 0; j < OUTC; ++j) s += __expf(r[j] - mx);
  float lse = mx + __logf(s);
#pragma unroll
  for (int j = 0; j < OUTC; ++j) row[j] = r[j] - lse;
}

extern "C" void kernel_launch(void* const* d_in, const int* in_sizes, int n_in,
                              void* d_out, int out_size, void* d_ws, size_t ws_size,
                              hipStream_t stream) {
  const float* x     = (const float*)d_in[0];
  const int*   edges = (const int*)d_in[1];      // [2, E] int32
  const float* W1    = (const float*)d_in[2];
  const float* b1    = (const float*)d_in[3];
  const float* W2    = (const float*)d_in[4];
  const float* b2    = (const float*)d_in[5];
  float* out = (float*)d_out;

  const long long N = (long long)in_sizes[0] / IN_DIM;   // 100000
  const long long E = (long long)in_sizes[1] / 2;        // 3200000

  // workspace layout (fp32): h1pre[N*16] | agg1[N*16] | h2pre[N*40]
  float* h1pre = (float*)d_ws;
  float* agg1  = h1pre + N * HID;
  float* h2pre = agg1  + N * HID;

  const int B = 256;

  // 1. agg1 = b1
  long long tot = N * HID;
  k_init_bias<<<(unsigned)((tot + B - 1) / B), B, 0, stream>>>(agg1, b1, HID, tot);

  // 2. h1pre = x @ W1   (fp32 WMMA)
  int ntiles = (int)(N / 16);
  long long thr = (long long)ntiles * 32;
  k_gemm1_wmma<<<(unsigned)((thr + B - 1) / B), B, 0, stream>>>(x, W1, h1pre, ntiles);

  // 3. scatter-add layer 1
  tot = E * HID;
  k_scatter16<<<(unsigned)((tot + B - 1) / B), B, 0, stream>>>(h1pre, agg1, edges, E);

  // 4. h2pre = relu(agg1) @ W2
  tot = N * OUTC;
  k_relu_gemm2<<<(unsigned)((tot + B - 1) / B), B, 0, stream>>>(agg1, W2, h2pre, N);

  // 5. out = b2
  k_init_bias<<<(unsigned)((tot + B - 1) / B), B, 0, stream>>>(out, b2, OUTC, tot);

  // 6. scatter-add layer 2 (directly into out)
  tot = E * OUTC;
  k_scatter40<<<(unsigned)((tot + B - 1) / B), B, 0, stream>>>(h2pre, out, edges, E);

  // 7. log_softmax in place
  k_log_softmax<<<(unsigned)((N + B - 1) / B), B, 0, stream>>>(out, N);
}